// SimpleS4_6313601925120
// MI455X (gfx1250) — compile-verified
//
#include <hip/hip_runtime.h>

// ---------------------------------------------------------------------------
// SimpleS4 collapsed via adjoint:
//   out_b = fc_b + sum_{t,i} x[b,t,i] * G[511-t, i]
//   G = S * B,  S[s,:] = (I+A)^s v,  v = C^T fc_w
// S built with log-depth Krylov doubling using f32 WMMA (16x16x4).
// Dominant cost: one streaming pass over x (268 MB ~ 11.5us at 23.3 TB/s).
// ---------------------------------------------------------------------------

typedef __attribute__((ext_vector_type(2))) float v2f;
typedef __attribute__((ext_vector_type(8))) float v8f;

// ---------------------------------------------------------------------------
// init: blocks [0,512)  -> M = A + I  (row per block)
//       blocks [512,528)-> v[h] = sum_i C[i,h]*fc_w[i]  -> S row 0
//       block 528       -> out[b] = fc_b (so reduce can just +=)
// ---------------------------------------------------------------------------
__global__ __launch_bounds__(512) void s4_init(
    const float* __restrict__ A, const float* __restrict__ Cm,
    const float* __restrict__ fc_w, const float* __restrict__ fc_b,
    float* __restrict__ M, float* __restrict__ S, float* __restrict__ out)
{
  int b = blockIdx.x;
  int tid = threadIdx.x;
  if (b < 512) {
    M[b * 512 + tid] = A[b * 512 + tid] + (b == tid ? 1.0f : 0.0f);
  } else if (b < 528) {
    __shared__ float red[512];
    int hh = tid & 31, slice = tid >> 5;        // 32 h-values x 16 i-slices
    int h = (b - 512) * 32 + hh;
    float acc = 0.0f;
    for (int i = slice; i < 512; i += 16)
      acc += Cm[i * 512 + h] * fc_w[i];
    red[tid] = acc;
    __syncthreads();
    if (slice == 0) {
      float s = 0.0f;
#pragma unroll
      for (int t = 0; t < 16; ++t) s += red[hh + 32 * t];
      S[h] = s;                                  // S row 0 = v
    }
  } else {
    if (tid < 256) out[tid] = fc_b[0];
  }
}

// ---------------------------------------------------------------------------
// One wave computes one 16x16 C tile of  C = A * B' (B' = B^T if transB),
// all matrices 512-wide row-major fp32, K = 512, via V_WMMA_F32_16X16X4_F32.
// A frag layout: lane(0-15|16-31) -> m=lane&15, VGPR r holds K = 2*(lane>>4)+r
// B frag layout: n=lane&15, VGPR r holds K = 2*(lane>>4)+r
// C/D layout:    VGPR r -> row tm*16 + 8*(lane>>4) + r, col tn*16 + (lane&15)
// ---------------------------------------------------------------------------
__device__ __forceinline__ void wmma_tile_f32(
    const float* A, const float* B, float* C, int tm, int tn, int transB)
{
  int lane = threadIdx.x & 31;
  int nm   = lane & 15;
  int half = lane >> 4;
  const float* aPtr = A + (size_t)(tm * 16 + nm) * 512 + 2 * half;
  v8f c = {};
  if (transB) {
    const float* bPtr = B + (size_t)(tn * 16 + nm) * 512 + 2 * half;
#pragma unroll 4
    for (int k0 = 0; k0 < 512; k0 += 4) {
      v2f a  = *(const v2f*)(aPtr + k0);
      v2f bb = *(const v2f*)(bPtr + k0);
      c = __builtin_amdgcn_wmma_f32_16x16x4_f32(false, a, false, bb,
                                                (short)0, c, false, false);
    }
  } else {
    const float* bPtr = B + (size_t)(2 * half) * 512 + tn * 16 + nm;
#pragma unroll 4
    for (int k0 = 0; k0 < 512; k0 += 4) {
      v2f a = *(const v2f*)(aPtr + k0);
      v2f bb;
      bb.x = bPtr[(size_t)k0 * 512];
      bb.y = bPtr[(size_t)k0 * 512 + 512];
      c = __builtin_amdgcn_wmma_f32_16x16x4_f32(false, a, false, bb,
                                                (short)0, c, false, false);
    }
  }
  float* cPtr = C + (size_t)(tm * 16 + half * 8) * 512 + tn * 16 + nm;
#pragma unroll
  for (int r = 0; r < 8; ++r) cPtr[(size_t)r * 512] = c[r];
}

// Two independent GEMMs in one launch (doubling step: apply + square).
// tiles = tilesM*32 (N fixed 512 -> 32 col tiles). Wave-uniform dispatch so
// EXEC stays all-ones for WMMA.
__global__ __launch_bounds__(128) void s4_dual_gemm(
    const float* A0, const float* B0, float* C0, int tiles0, int tB0,
    const float* A1, const float* B1, float* C1, int tiles1, int tB1)
{
  int w = blockIdx.x * 4 + (int)(threadIdx.x >> 5);
  const float *A, *B;
  float* C;
  int tB, t;
  if (w < tiles0) {
    A = A0; B = B0; C = C0; tB = tB0; t = w;
  } else {
    int w1 = w - tiles0;
    if (w1 >= tiles1) return;
    A = A1; B = B1; C = C1; tB = tB1; t = w1;
  }
  wmma_tile_f32(A, B, C, t >> 5, t & 31, tB);
}

// ---------------------------------------------------------------------------
// Streaming reduction: out[b] += sum_{t,i} x[b,t,i] * G[511-t, i]
// One block per batch row; float4 loads; deterministic LDS tree reduce.
// ---------------------------------------------------------------------------
__global__ __launch_bounds__(512) void s4_reduce(
    const float* __restrict__ x, const float* __restrict__ G,
    float* __restrict__ out)
{
  int b = blockIdx.x;
  int tid = threadIdx.x;
  const float4* X4 = (const float4*)(x + (size_t)b * 262144);
  float acc = 0.0f;
  for (int j4 = tid; j4 < 65536; j4 += 512) {
    float4 xv = X4[j4];
    int t  = j4 >> 7;                 // 128 float4 per 512-wide row
    int i4 = j4 & 127;
    const float4* G4 = (const float4*)(G + (size_t)(511 - t) * 512);
    float4 gv = G4[i4];
    acc += xv.x * gv.x + xv.y * gv.y + xv.z * gv.z + xv.w * gv.w;
  }
  __shared__ float red[512];
  red[tid] = acc;
  __syncthreads();
  for (int s = 256; s > 0; s >>= 1) {
    if (tid < s) red[tid] += red[tid + s];
    __syncthreads();
  }
  if (tid == 0) out[b] += red[0];
}

// ---------------------------------------------------------------------------
extern "C" void kernel_launch(void* const* d_in, const int* in_sizes, int n_in,
                              void* d_out, int out_size, void* d_ws, size_t ws_size,
                              hipStream_t stream)
{
  const float* x    = (const float*)d_in[0];   // [256,512,512]
  const float* A    = (const float*)d_in[1];   // [512,512]
  const float* Bm   = (const float*)d_in[2];   // [512,512]
  const float* Cm   = (const float*)d_in[3];   // [512,512]
  const float* fc_w = (const float*)d_in[4];   // [512]
  const float* fc_b = (const float*)d_in[5];   // [1]
  float* out = (float*)d_out;                  // [256]

  float* S  = (float*)d_ws;                    // 512x512 Krylov rows M^s v
  float* G  = S  + 512 * 512;                  // 512x512  G = S*B
  float* P0 = G  + 512 * 512;                  // M power ping
  float* P1 = P0 + 512 * 512;                  // M power pong

  // M = A + I, S[0] = C^T fc_w, out = fc_b
  s4_init<<<529, 512, 0, stream>>>(A, Cm, fc_w, fc_b, P0, S, out);

  // Krylov doubling: step k writes S rows [2^k, 2^{k+1}) = S[0:2^k) * P^T
  // and (fused) squares P -> M^{2^{k+1}} for the next step.
  for (int k = 0; k < 9; ++k) {
    float* Pin  = (k & 1) ? P1 : P0;
    float* Pout = (k & 1) ? P0 : P1;
    int rows     = 1 << k;
    int tilesApp = ((rows + 15) / 16) * 32;
    int tilesSq  = (k < 8) ? 1024 : 0;          // last power is unused
    int total    = tilesApp + tilesSq;
    s4_dual_gemm<<<(total + 3) / 4, 128, 0, stream>>>(
        S, Pin, S + (size_t)rows * 512, tilesApp, /*transB=*/1,
        Pin, Pin, Pout, tilesSq, /*transB=*/0);
  }

  // G[s,i] = sum_h S[s,h] * B[h,i]
  s4_dual_gemm<<<256, 128, 0, stream>>>(S, Bm, G, 1024, 0,
                                        nullptr, nullptr, nullptr, 0, 0);

  // out[b] += <x[b], G flipped in t>
  s4_reduce<<<256, 512, 0, stream>>>(x, G, out);
}